// SubjectMapLayer_84593675862103
// MI455X (gfx1250) — compile-verified
//
#include <hip/hip_runtime.h>
#include <math.h>

// ---------------- static problem configuration (matches reference) ----------
#define NCH   256                 // N
#define NMODE 8                   // M
#define NSUBJ 64                  // S
#define VECL  32896               // N*(N+1)/2
#define DIAG_SHIFT 1.0e-5f
#define EPS_DIAG   1.0e-6f
#define LDP   257                 // padded LDS pitch for Cholesky (257KB < 320KB WGP LDS)

typedef __attribute__((ext_vector_type(2))) float v2f;
typedef __attribute__((ext_vector_type(8))) float v8f;

__device__ __forceinline__ float softplusf(float x) {
    return (x > 0.0f) ? (x + log1pf(expf(-x))) : log1pf(expf(x));
}
__device__ __forceinline__ float softplus_invf(float y) {
    // log(expm1(y)) stable for y > 0 (Cholesky diag of SPD + identity => y ~ 1)
    return y + logf(-expm1f(-y));
}

// ---------------------------------------------------------------------------
// Kernel A: per-mode Cholesky of group_map in LDS, diag -> softplus_inverse,
// write zero-padded lower-triangular gL[m] (NCH x NCH) to workspace.
// One workgroup (256 threads) per mode; dynamic LDS 256*257 floats.
// ---------------------------------------------------------------------------
extern "C" __global__ __launch_bounds__(NCH)
void group_cholesky_kernel(const float* __restrict__ gm, float* __restrict__ gL) {
    extern __shared__ float W[];                 // [NCH][LDP]
    const int m   = blockIdx.x;
    const int tid = threadIdx.x;                 // row owned by this thread
    const float* A = gm + (size_t)m * NCH * NCH;

    for (int r = 0; r < NCH; ++r)                // coalesced load
        W[r * LDP + tid] = A[r * NCH + tid];
    __syncthreads();

    for (int k = 0; k < NCH; ++k) {
        if (tid == 0) W[k * LDP + k] = sqrtf(W[k * LDP + k]);
        __syncthreads();
        const float piv = W[k * LDP + k];
        if (tid > k) W[tid * LDP + k] /= piv;
        __syncthreads();
        if (tid > k) {
            const float lik = W[tid * LDP + k];
            for (int j = k + 1; j <= tid; ++j)   // trailing rank-1 update, row tid
                W[tid * LDP + j] = fmaf(-lik, W[j * LDP + k], W[tid * LDP + j]);
        }
        __syncthreads();
    }

    // Replace diagonal with unconstrained parameter (FillScaleTriL.inverse)
    const float d = softplus_invf(W[tid * LDP + tid] - DIAG_SHIFT);
    __syncthreads();
    W[tid * LDP + tid] = d;
    __syncthreads();

    float* out = gL + (size_t)m * NCH * NCH;     // coalesced store, zero above diag
    for (int r = 0; r < NCH; ++r) {
        const float v = (tid <= r) ? W[r * LDP + tid] : 0.0f;
        out[r * NCH + tid] = v;
    }
}

// ---------------------------------------------------------------------------
// Kernel B: per (subject,mode) SYRK  C = L * L^T + eps*I  via WMMA fp32.
// L[r][c] = gL[m][r][c] + dev[s][m][perm(r,c)]  (r>c),
// L[r][r] = softplus(gL_diag + dev) + 1e-5, 0 above diagonal.
// perm(r,c): TFP fill_triangular index map, closed form.
// Only lower 64x64 tiles computed (10 of 16); mirrored on store.
// ---------------------------------------------------------------------------
#define TILE   64
#define LDS_LD 68

__device__ __forceinline__ void fill_panel(float* __restrict__ buf,
                                           const float* __restrict__ gLm,
                                           const float* __restrict__ devv,
                                           int rowBase, int colBase) {
    const int tid = threadIdx.x;
    for (int idx = tid; idx < TILE * TILE; idx += 256) {
        const int pr = idx >> 6, pc = idx & 63;
        const int r = rowBase + pr, c = colBase + pc;
        float v = 0.0f;
        if (c <= r) {
            const int f = r * NCH + c;
            const int p = (f < (VECL - NCH)) ? (NCH + f) : (2 * VECL - NCH - 1 - f);
            const float x = gLm[f] + devv[p];
            v = (c == r) ? (softplusf(x) + DIAG_SHIFT) : x;
        }
        buf[pr * LDS_LD + pc] = v;               // contiguous pc => coalesced dev reads
    }
}

extern "C" __global__ __launch_bounds__(256)
void subject_syrk_kernel(const float* __restrict__ gL,
                         const float* __restrict__ dev,
                         float* __restrict__ out) {
    __shared__ float Abuf[TILE * LDS_LD];
    __shared__ float Bbuf[TILE * LDS_LD];

    const int sm = blockIdx.y;                   // s*8 + m
    const int m  = sm & 7;
    // map blockIdx.x in [0,10) -> lower-triangular 64x64 tile (bi >= bj)
    int t = blockIdx.x, bi = 0;
    while (t >= bi + 1) { t -= bi + 1; ++bi; }
    const int bj = t;

    const float* gLm  = gL  + (size_t)m  * NCH * NCH;
    const float* devv = dev + (size_t)sm * VECL;

    const int lane = threadIdx.x & 31;
    const int wave = threadIdx.x >> 5;
    const int ln16 = lane & 15;                  // matrix row (A) / col (B,C) index
    const int kb   = (lane >> 4) << 1;           // K sub-offset per ISA A-layout
    // each wave owns two 16x16 sub-tiles of the 64x64 C tile
    const int tt0 = wave, tt1 = wave + 8;
    const int tm0 = (tt0 >> 2) * 16, tn0 = (tt0 & 3) * 16;
    const int tm1 = (tt1 >> 2) * 16, tn1 = (tt1 & 3) * 16;

    v8f acc0 = {};
    v8f acc1 = {};

    for (int kt = 0; kt <= bj; ++kt) {           // L lower-tri: K panels 0..bj only
        fill_panel(Abuf, gLm, devv, bi * TILE, kt * TILE);
        if (bi != bj) fill_panel(Bbuf, gLm, devv, bj * TILE, kt * TILE);
        __syncthreads();
        const float* Bp = (bi == bj) ? Abuf : Bbuf;

        #pragma unroll
        for (int k = 0; k < TILE; k += 4) {
            v2f a0, b0, a1, b1;
            a0.x = Abuf[(tm0 + ln16) * LDS_LD + k + kb];
            a0.y = Abuf[(tm0 + ln16) * LDS_LD + k + kb + 1];
            b0.x = Bp  [(tn0 + ln16) * LDS_LD + k + kb];      // B = L_J^T
            b0.y = Bp  [(tn0 + ln16) * LDS_LD + k + kb + 1];
            a1.x = Abuf[(tm1 + ln16) * LDS_LD + k + kb];
            a1.y = Abuf[(tm1 + ln16) * LDS_LD + k + kb + 1];
            b1.x = Bp  [(tn1 + ln16) * LDS_LD + k + kb];
            b1.y = Bp  [(tn1 + ln16) * LDS_LD + k + kb + 1];
            acc0 = __builtin_amdgcn_wmma_f32_16x16x4_f32(false, a0, false, b0,
                                                         (short)0, acc0, false, false);
            acc1 = __builtin_amdgcn_wmma_f32_16x16x4_f32(false, a1, false, b1,
                                                         (short)0, acc1, false, false);
        }
        __syncthreads();
    }

    // store C (and mirror): C layout VGPR v -> M = v + 8*(lane/16), N = lane%16
    float* outM = out + (size_t)sm * NCH * NCH;
    const int half8 = (lane >> 4) << 3;
    #pragma unroll
    for (int v = 0; v < 8; ++v) {
        {
            const int row = bi * TILE + tm0 + v + half8;
            const int col = bj * TILE + tn0 + ln16;
            const float val = acc0[v] + ((row == col) ? EPS_DIAG : 0.0f);
            outM[row * NCH + col] = val;
            if (bi != bj) outM[col * NCH + row] = val;
        }
        {
            const int row = bi * TILE + tm1 + v + half8;
            const int col = bj * TILE + tn1 + ln16;
            const float val = acc1[v] + ((row == col) ? EPS_DIAG : 0.0f);
            outM[row * NCH + col] = val;
            if (bi != bj) outM[col * NCH + row] = val;
        }
    }
}

// ---------------------------------------------------------------------------
extern "C" void kernel_launch(void* const* d_in, const int* in_sizes, int n_in,
                              void* d_out, int out_size, void* d_ws, size_t ws_size,
                              hipStream_t stream) {
    (void)in_sizes; (void)n_in; (void)out_size; (void)ws_size;
    const float* gm  = (const float*)d_in[0];   // (8, 256, 256)
    const float* dev = (const float*)d_in[1];   // (64, 8, 32896)
    float* out = (float*)d_out;                 // (64, 8, 256, 256)
    float* gL  = (float*)d_ws;                  // 8*256*256 floats = 2 MB scratch

    const size_t cholLds = (size_t)NCH * LDP * sizeof(float);   // 257 KB (<320 KB)
    hipLaunchKernelGGL(group_cholesky_kernel, dim3(NMODE), dim3(NCH), cholLds, stream,
                       gm, gL);
    hipLaunchKernelGGL(subject_syrk_kernel, dim3(10, NSUBJ * NMODE), dim3(256), 0, stream,
                       gL, dev, out);
}